// Whitening2d_25546465477197
// MI455X (gfx1250) — compile-verified
//
#include <hip/hip_runtime.h>
#include <hip/hip_bf16.h>

typedef __attribute__((ext_vector_type(2))) float v2f;
typedef __attribute__((ext_vector_type(4))) float v4f;
typedef __attribute__((ext_vector_type(8))) float v8f;
typedef __attribute__((ext_vector_type(4))) int   v4i;

#define NROWS   524288
#define CH      128
#define PITCH   132          // LDS row pitch in floats (bank-conflict padding, 16B aligned)
#define NCOPIES 128          // partial-Gram copies == pass1 grid size
#define SLAB    128          // rows per LDS slab
#define SLABF   (SLAB * PITCH)   // floats per slab buffer
#define SLABS_PER_BLK 32     // 128 blocks * 32 slabs * 128 rows = 524288
#define GEMM_ROWS 128
#define GEMM_BLOCKS 512
#define GEMM_SLABS 8         // 512 blocks * 8 slabs * 128 rows = 524288

// workspace layout (float offsets)
#define WS_WT    0                         // 128x128  W^T (row k = input ch, col o = out ch)
#define WS_BIAS  (WS_WT + CH*CH)           // 128      bias = W @ mean
#define WS_MEAN  (WS_BIAS + CH)            // 128
#define WS_COV   (WS_MEAN + CH)            // 128x128
#define WS_SUMSP (WS_COV + CH*CH)          // NCOPIES x 128 partial column sums
#define WS_GRAMP (WS_SUMSP + NCOPIES*CH)   // NCOPIES x 128x128 partial Gram
// total ~8.2 MiB of f32 scratch

// ---------------------------------------------------------------------------
// Async global->LDS 16B copy (gfx1250 GLOBAL_LOAD_ASYNC_TO_LDS_B128, ASYNCcnt
// tracked). Builtin signature (from hipcc diagnostic): param0 = AS1 v4i*,
// param1 = AS3 v4i*, then imm offset, imm cpol. Falls back to a synchronous
// copy if the builtin is unavailable.
// ---------------------------------------------------------------------------
#if __has_builtin(__builtin_amdgcn_global_load_async_to_lds_b128)
#define HAS_ASYNC_LDS 1
#else
#define HAS_ASYNC_LDS 0
#endif

__device__ __forceinline__ void copy16_g2l(const float* g, float* l) {
#if HAS_ASYNC_LDS
    __builtin_amdgcn_global_load_async_to_lds_b128(
        (__attribute__((address_space(1))) v4i*)(void*)g,
        (__attribute__((address_space(3))) v4i*)(void*)l, 0, 0);
#else
    *(v4f*)l = *(const v4f*)g;
#endif
}

__device__ __forceinline__ void async_wait0() {
#if HAS_ASYNC_LDS
#if __has_builtin(__builtin_amdgcn_s_wait_asynccnt)
    __builtin_amdgcn_s_wait_asynccnt(0);
#else
    asm volatile("s_wait_asynccnt 0x0" ::: "memory");
#endif
#endif
}

// cooperative 128x128 slab load: 512 threads x 8 x 16B, fully coalesced
__device__ __forceinline__ void issue_slab(const float* xsrc, float* xs, int tid) {
    #pragma unroll
    for (int i = 0; i < 8; ++i) {
        int idx4 = tid + i * 512;              // 0..4095
        int r    = idx4 >> 5;
        int c4   = (idx4 & 31) << 2;
        copy16_g2l(xsrc + r * CH + c4, xs + r * PITCH + c4);
    }
}

// ---------------------------------------------------------------------------
// Pass 1: per-block partial Gram (X^T X) via f32 WMMA + partial column sums.
// Double-buffered async slab loads; each block owns one private output copy
// -> no atomics, deterministic.
// ---------------------------------------------------------------------------
__global__ __launch_bounds__(512)
void pass1_gram_sums(const float* __restrict__ x, float* __restrict__ ws) {
    extern __shared__ float smem[];
    float* sred = smem + 2 * SLABF;            // 4 * CH

    const int tid  = threadIdx.x;
    const int lane = tid & 31;
    const int wave = tid >> 5;                 // 0..15
    const int blk  = blockIdx.x;               // 0..NCOPIES-1
    const long rowbase0 = (long)blk * (SLAB * (long)SLABS_PER_BLK);

    const int itile = wave >> 1;               // 0..7  (Gram row tile)
    const int jbase = (wave & 1) * 4;          // j tiles jbase..jbase+3
    const int khl   = (lane >> 4) << 1;        // K sub-offset (0 or 2) per ISA layout
    const int ml    = lane & 15;               // M (A) / N (B) within tile

    v8f acc[4] = {};
    float colsum = 0.f;
    const int sc = tid & 127;                  // channel this thread sums
    const int sq = tid >> 7;                   // row-quarter 0..3

    issue_slab(x + rowbase0 * CH, smem, tid);  // slab 0 -> buffer 0

    for (int s = 0; s < SLABS_PER_BLK; ++s) {
        async_wait0();
        __syncthreads();                       // slab s resident, all waves past slab s-1
        if (s + 1 < SLABS_PER_BLK)
            issue_slab(x + (rowbase0 + (long)(s + 1) * SLAB) * CH,
                       smem + ((s + 1) & 1) * SLABF, tid);   // prefetch overlaps compute
        const float* xs = smem + (s & 1) * SLABF;

        // column partial sums (each thread: 32 rows of one channel)
        #pragma unroll 8
        for (int r = 0; r < 32; ++r)
            colsum += xs[(sq * 32 + r) * PITCH + sc];

        // Gram tiles: D += A(16x4) * B(4x16), K = slab rows
        for (int kb = 0; kb < SLAB; kb += 4) {
            v2f a;
            a.x = xs[(kb + khl    ) * PITCH + itile * 16 + ml];
            a.y = xs[(kb + khl + 1) * PITCH + itile * 16 + ml];
            #pragma unroll
            for (int j = 0; j < 4; ++j) {
                v2f b;
                b.x = xs[(kb + khl    ) * PITCH + (jbase + j) * 16 + ml];
                b.y = xs[(kb + khl + 1) * PITCH + (jbase + j) * 16 + ml];
                acc[j] = __builtin_amdgcn_wmma_f32_16x16x4_f32(
                    false, a, false, b, (short)0, acc[j], false, false);
            }
        }
        __syncthreads();
    }

    // reduce the 4 row-quarters of the column sums, write this block's copy
    sred[sq * CH + sc] = colsum;
    __syncthreads();
    if (tid < CH) {
        float s = sred[tid] + sred[CH + tid] + sred[2 * CH + tid] + sred[3 * CH + tid];
        ws[WS_SUMSP + blk * CH + tid] = s;
    }

    // write this block's private Gram partial (C/D layout: lane%16=N, vgpr v=row)
    float* gp = ws + WS_GRAMP + (size_t)blk * (CH * CH);
    const int rhalf = (lane >> 4) * 8;
    #pragma unroll
    for (int j = 0; j < 4; ++j) {
        int col = (jbase + j) * 16 + ml;
        #pragma unroll
        for (int v = 0; v < 8; ++v) {
            int row = itile * 16 + rhalf + v;
            gp[row * CH + col] = acc[j][v];
        }
    }
}

// ---------------------------------------------------------------------------
// Deterministic serial reductions over the NCOPIES partial copies.
// ---------------------------------------------------------------------------
__global__ __launch_bounds__(128)
void reduce_sums(float* __restrict__ ws) {
    int c = threadIdx.x;
    float s = 0.f;
    for (int b = 0; b < NCOPIES; ++b) s += ws[WS_SUMSP + b * CH + c];
    ws[WS_MEAN + c] = s * (1.0f / (float)NROWS);
}

__global__ __launch_bounds__(256)
void reduce_gram(float* __restrict__ ws) {
    int e = blockIdx.x * 256 + threadIdx.x;   // < CH*CH
    float g = 0.f;
    for (int b = 0; b < NCOPIES; ++b) g += ws[WS_GRAMP + (size_t)b * (CH * CH) + e];
    int i = e >> 7, j = e & 127;
    float mi = ws[WS_MEAN + i], mj = ws[WS_MEAN + j];
    ws[WS_COV + e] = (g - (float)NROWS * mi * mj) * (1.0f / (float)(NROWS - 1));
}

// ---------------------------------------------------------------------------
// Single-workgroup Cholesky + triangular inverse (W = L^{-1}), writes
// W^T (for conflict-free B loads in the GEMM) and bias = W @ mean.
// ---------------------------------------------------------------------------
__global__ __launch_bounds__(128)
void chol_inv(float* __restrict__ ws) {
    extern __shared__ float smem[];
    float* A = smem;             // 128x128
    float* W = smem + CH * CH;   // 128x128 (lower-tri inverse)
    const int t = threadIdx.x;

    for (int j = 0; j < CH; ++j) A[t * CH + j] = ws[WS_COV + t * CH + j];
    __syncthreads();

    // right-looking Cholesky, one row per thread
    for (int k = 0; k < CH; ++k) {
        if (t == k) A[k * CH + k] = sqrtf(A[k * CH + k]);
        __syncthreads();
        if (t > k) A[t * CH + k] /= A[k * CH + k];
        __syncthreads();
        if (t > k) {
            float lik = A[t * CH + k];
            for (int j = k + 1; j <= t; ++j) A[t * CH + j] -= lik * A[j * CH + k];
        }
        __syncthreads();
    }

    // forward solve L * w = e_t  (thread t owns column t of W)
    for (int i = t; i < CH; ++i) {
        float s = (i == t) ? 1.0f : 0.0f;
        for (int p = t; p < i; ++p) s -= A[i * CH + p] * W[p * CH + t];
        W[i * CH + t] = s / A[i * CH + i];
    }
    __syncthreads();

    // bias[o] = sum_i W[o][i] * mean[i]
    float b = 0.f;
    for (int i = 0; i <= t; ++i) b += W[t * CH + i] * ws[WS_MEAN + i];
    ws[WS_BIAS + t] = b;

    // W^T: row k = input channel, col o = output channel; zero strict upper of W
    for (int k = 0; k < CH; ++k)
        ws[WS_WT + k * CH + t] = (k <= t) ? W[t * CH + k] : 0.0f;
}

// ---------------------------------------------------------------------------
// Y = X @ W^T - bias via f32 WMMA. Persistent blocks: W^T loaded once per
// block (L2 traffic /8), x slabs double-buffered with async loads.
// ---------------------------------------------------------------------------
__global__ __launch_bounds__(512)
void gemm_out(const float* __restrict__ x, const float* __restrict__ ws,
              float* __restrict__ y) {
    extern __shared__ float smem[];
    float* wt = smem + 2 * SLABF;              // 128x128 W^T
    float* bs = wt + CH * CH;                  // 128 bias

    const int tid  = threadIdx.x;
    const int lane = tid & 31;
    const int wave = tid >> 5;                 // 0..15
    const long blkrow0 = (long)blockIdx.x * (GEMM_SLABS * (long)GEMM_ROWS);

    const int ntile = wave & 7;                // output-channel tile
    const int mbase = (wave >> 3) * 4;         // row tiles mbase..mbase+3
    const int khl   = (lane >> 4) << 1;
    const int ml    = lane & 15;

    // prologue: W^T (4096 x 16B), bias (32 x 16B), first slab — all async
    #pragma unroll
    for (int i = 0; i < 8; ++i) {
        int idx4 = (tid + i * 512) << 2;
        copy16_g2l(ws + WS_WT + idx4, wt + idx4);
    }
    if (tid < 32) copy16_g2l(ws + WS_BIAS + tid * 4, bs + tid * 4);
    issue_slab(x + blkrow0 * CH, smem, tid);   // slab 0 -> buffer 0

    for (int s = 0; s < GEMM_SLABS; ++s) {
        async_wait0();
        __syncthreads();
        if (s + 1 < GEMM_SLABS)
            issue_slab(x + (blkrow0 + (long)(s + 1) * SLAB) * CH,
                       smem + ((s + 1) & 1) * SLABF, tid);
        const float* xs = smem + (s & 1) * SLABF;

        v8f acc[4] = {};
        for (int kb = 0; kb < CH; kb += 4) {
            v2f b;                             // B[k][n] = W^T[k][n0+n], lane-consecutive
            b.x = wt[(kb + khl    ) * CH + ntile * 16 + ml];
            b.y = wt[(kb + khl + 1) * CH + ntile * 16 + ml];
            #pragma unroll
            for (int m = 0; m < 4; ++m) {
                int mrow = (mbase + m) * 16 + ml;
                v2f a;                         // A[m][k] = xs[row][k]
                a.x = xs[mrow * PITCH + kb + khl];
                a.y = xs[mrow * PITCH + kb + khl + 1];
                acc[m] = __builtin_amdgcn_wmma_f32_16x16x4_f32(
                    false, a, false, b, (short)0, acc[m], false, false);
            }
        }

        const float bias = bs[ntile * 16 + ml];
        const long rowbase = blkrow0 + (long)s * GEMM_ROWS;
        const int rhalf = (lane >> 4) * 8;
        #pragma unroll
        for (int m = 0; m < 4; ++m) {
            #pragma unroll
            for (int v = 0; v < 8; ++v) {
                long row = rowbase + (mbase + m) * 16 + rhalf + v;
                y[row * CH + ntile * 16 + ml] = acc[m][v] - bias;
            }
        }
        __syncthreads();                       // acc stored; safe to overwrite xs next iter
    }
}

// ---------------------------------------------------------------------------
extern "C" void kernel_launch(void* const* d_in, const int* in_sizes, int n_in,
                              void* d_out, int out_size, void* d_ws, size_t ws_size,
                              hipStream_t stream) {
    (void)in_sizes; (void)n_in; (void)out_size; (void)ws_size;
    const float* x = (const float*)d_in[0];
    float* y  = (float*)d_out;
    float* ws = (float*)d_ws;

    const size_t lds_pass1 = (2 * SLABF + 4 * CH) * sizeof(float);            // ~137 KB
    const size_t lds_chol  = (2 * CH * CH) * sizeof(float);                   // 128 KB
    const size_t lds_gemm  = (2 * SLABF + CH * CH + CH) * sizeof(float);      // ~201 KB

    pass1_gram_sums<<<NCOPIES, 512, lds_pass1, stream>>>(x, ws);
    reduce_sums<<<1, 128, 0, stream>>>(ws);
    reduce_gram<<<(CH * CH) / 256, 256, 0, stream>>>(ws);
    chol_inv<<<1, 128, lds_chol, stream>>>(ws);
    gemm_out<<<GEMM_BLOCKS, 512, lds_gemm, stream>>>(x, ws, y);
}